// CrossEntropyLoss_27539330302301
// MI455X (gfx1250) — compile-verified
//
#include <hip/hip_runtime.h>
#include <stdint.h>

#define ROWS 8192
#define COLS 32000
#define TPB 320                 // 10 waves of 32 (wave32)
#define VEC 4
#define CHUNK (TPB * VEC)       // 1280 floats per block-iteration
#define NITER (COLS / CHUNK)    // 25, exact
#define CHUNK_BYTES (CHUNK * 4) // 5120 B
#define PIPE 4                  // async pipeline depth (chunks in flight)

// Combine two online-softmax partial states (m,s): numerically safe because
// identity uses m=-FLT_MAX (finite), so exp() underflows to 0 instead of NaN.
__device__ __forceinline__ void combine_ms(float& m, float& s, float om, float os) {
    float nm = fmaxf(m, om);
    s = s * __expf(m - nm) + os * __expf(om - nm);
    m = nm;
}

__device__ __forceinline__ void async_copy16(unsigned lds_addr, unsigned voff,
                                             const float* sbase) {
    asm volatile("global_load_async_to_lds_b128 %0, %1, %2"
                 :: "v"(lds_addr), "v"(voff), "s"(sbase) : "memory");
}

__global__ __launch_bounds__(TPB) void ce_row_kernel(const float* __restrict__ pred,
                                                     const float* __restrict__ tgt,
                                                     float* __restrict__ row_loss) {
    __shared__ __align__(16) float shp[PIPE][CHUNK];   // pred staging, 4-deep
    __shared__ __align__(16) float sht[PIPE][CHUNK];   // target staging, 4-deep
    __shared__ float red[4][16];                       // cross-wave partials

    const int tid = threadIdx.x;
    const int row = blockIdx.x;
    const float* prow = pred + (size_t)row * COLS;
    const float* trow = tgt  + (size_t)row * COLS;

    // LDS byte offsets of this lane's 16B slot in each pipeline stage.
    const unsigned lane16 = (unsigned)tid * 16u;
    unsigned ldsP[PIPE], ldsT[PIPE];
#pragma unroll
    for (int b = 0; b < PIPE; ++b) {
        ldsP[b] = (unsigned)(uintptr_t)(&shp[b][0]) + lane16;
        ldsT[b] = (unsigned)(uintptr_t)(&sht[b][0]) + lane16;
    }

    // Prologue: issue chunks 0..PIPE-2 (async global -> LDS, ASYNCcnt).
#pragma unroll
    for (int c = 0; c < PIPE - 1; ++c) {
        unsigned voff = (unsigned)c * (unsigned)CHUNK_BYTES + lane16;
        async_copy16(ldsP[c], voff, prow);
        async_copy16(ldsT[c], voff, trow);
    }

    float m = -3.0e38f, s = 0.0f, tsum = 0.0f, dsum = 0.0f;

    for (int i = 0; i < NITER; ++i) {
        const int cur = i & (PIPE - 1);

        // Keep PIPE chunks in flight.
        if (i + PIPE - 1 < NITER) {
            const int nb = (i + PIPE - 1) & (PIPE - 1);
            unsigned voff = (unsigned)(i + PIPE - 1) * (unsigned)CHUNK_BYTES + lane16;
            async_copy16(ldsP[nb], voff, prow);
            async_copy16(ldsT[nb], voff, trow);
        }

        // Retire chunk i: async ops complete in order per wave, so waiting for
        // <= 2*(#chunks still wanted in flight) guarantees chunk i landed.
        const int rem = NITER - 1 - i;  // chunks beyond i still pending
        if (rem >= PIPE - 1)      asm volatile("s_wait_asynccnt 6" ::: "memory");
        else if (rem == 2)        asm volatile("s_wait_asynccnt 4" ::: "memory");
        else if (rem == 1)        asm volatile("s_wait_asynccnt 2" ::: "memory");
        else                      asm volatile("s_wait_asynccnt 0" ::: "memory");

        const float4 p = *reinterpret_cast<const float4*>(&shp[cur][tid * VEC]);
        const float4 t = *reinterpret_cast<const float4*>(&sht[cur][tid * VEC]);

        // Online softmax accumulation (1 exp/element steady state).
        float cm = fmaxf(fmaxf(p.x, p.y), fmaxf(p.z, p.w));
        if (cm > m) { s *= __expf(m - cm); m = cm; }
        s += __expf(p.x - m) + __expf(p.y - m) + __expf(p.z - m) + __expf(p.w - m);

        tsum += (t.x + t.y) + (t.z + t.w);
        dsum += p.x * t.x + p.y * t.y + p.z * t.z + p.w * t.w;
    }

    // Wave32 butterfly reduction.
    for (int off = 16; off > 0; off >>= 1) {
        float om = __shfl_xor(m, off);
        float os = __shfl_xor(s, off);
        combine_ms(m, s, om, os);
        tsum += __shfl_xor(tsum, off);
        dsum += __shfl_xor(dsum, off);
    }

    const int wave = tid >> 5, lane = tid & 31;
    if (lane == 0) {
        red[0][wave] = m; red[1][wave] = s; red[2][wave] = tsum; red[3][wave] = dsum;
    }
    __syncthreads();

    if (wave == 0) {
        const int nw = TPB / 32; // 10
        m    = (lane < nw) ? red[0][lane] : -3.0e38f;
        s    = (lane < nw) ? red[1][lane] : 0.0f;
        tsum = (lane < nw) ? red[2][lane] : 0.0f;
        dsum = (lane < nw) ? red[3][lane] : 0.0f;
        for (int off = 16; off > 0; off >>= 1) {
            float om = __shfl_xor(m, off);
            float os = __shfl_xor(s, off);
            combine_ms(m, s, om, os);
            tsum += __shfl_xor(tsum, off);
            dsum += __shfl_xor(dsum, off);
        }
        if (lane == 0) {
            float lse = m + __logf(s);
            row_loss[row] = tsum * lse - dsum;   // Σt·lse − Σp·t
        }
    }
}

__global__ __launch_bounds__(256) void ce_mean_kernel(const float* __restrict__ row_loss,
                                                      float* __restrict__ out) {
    __shared__ float sm[8];
    float acc = 0.0f;
    for (int i = threadIdx.x; i < ROWS; i += 256) acc += row_loss[i];
    for (int off = 16; off > 0; off >>= 1) acc += __shfl_xor(acc, off);
    const int wave = threadIdx.x >> 5, lane = threadIdx.x & 31;
    if (lane == 0) sm[wave] = acc;
    __syncthreads();
    if (wave == 0) {
        acc = (lane < 8) ? sm[lane] : 0.0f;
        for (int off = 4; off > 0; off >>= 1) acc += __shfl_xor(acc, off);
        if (lane == 0) out[0] = acc * (1.0f / (float)ROWS);
    }
}

extern "C" void kernel_launch(void* const* d_in, const int* in_sizes, int n_in,
                              void* d_out, int out_size, void* d_ws, size_t ws_size,
                              hipStream_t stream) {
    (void)in_sizes; (void)n_in; (void)out_size; (void)ws_size;
    const float* pred = (const float*)d_in[0];
    const float* tgt  = (const float*)d_in[1];
    float* row_loss = (float*)d_ws;          // 8192 floats
    float* out      = (float*)d_out;         // 1 float

    ce_row_kernel<<<ROWS, TPB, 0, stream>>>(pred, tgt, row_loss);
    ce_mean_kernel<<<1, 256, 0, stream>>>(row_loss, out);
}